// TabRMv3_53721450939152
// MI455X (gfx1250) — compile-verified
//
#include <hip/hip_runtime.h>

typedef __attribute__((ext_vector_type(2))) float v2f;
typedef __attribute__((ext_vector_type(8))) float v8f;

#define N_TOK 8192
#define Dm 256
#define Hm 64
#define Em 32
#define MAXTILES 544      // >= sum_e ceil(count_e/16) worst case (512 + 32)
#define SA_STRIDE 260     // 260 % 64 == 4 -> conflict-free column reads, 16B aligned
#define SH_STRIDE 68      // 68 % 64 == 4

// ---------------- kernel 0: zero expert counts ----------------
__global__ void moe_zero_counts(int* __restrict__ counts) {
    if (threadIdx.x < Em) counts[threadIdx.x] = 0;
}

// ---------------- kernel 1: router logits via f32 WMMA + wave argmax + counts ----
// One wave32 per 16-token tile. logits tile = X[16x256] @ RW^T[256x32]
// N=32 experts -> two 16-col WMMA accumulator tiles (c0: experts 0-15, c1: 16-31).
// B = RW^T means the (k,k+1) fragment pair for one expert column is CONTIGUOUS
// in router_w -> single 8B loads.
__global__ __launch_bounds__(128) void moe_router(const float* __restrict__ x,
                                                  const float* __restrict__ rw,
                                                  const float* __restrict__ rb,
                                                  int* __restrict__ idx,
                                                  int* __restrict__ counts) {
    int wave = threadIdx.x >> 5;
    int lane = threadIdx.x & 31;
    int half = lane >> 4;       // 0: lanes 0-15, 1: lanes 16-31
    int m    = lane & 15;       // A row within half / B-C column within tile
    int kb   = half * 2;        // K sub-offset per ISA f32 A/B layout
    int n0   = (blockIdx.x * 4 + wave) * 16;   // token-tile base

    const float* xr = x  + (size_t)(n0 + m) * Dm;
    const float* w0 = rw + (size_t)m * Dm;          // expert m    (tile 0)
    const float* w1 = rw + (size_t)(m + 16) * Dm;   // expert m+16 (tile 1)

    v8f c0 = {0.f, 0.f, 0.f, 0.f, 0.f, 0.f, 0.f, 0.f};
    v8f c1 = {0.f, 0.f, 0.f, 0.f, 0.f, 0.f, 0.f, 0.f};
    for (int k0 = 0; k0 < Dm; k0 += 4) {
        float2 av  = *(const float2*)(xr + k0 + kb);
        float2 bv0 = *(const float2*)(w0 + k0 + kb);
        float2 bv1 = *(const float2*)(w1 + k0 + kb);
        v2f a, b0, b1;
        a.x = av.x;   a.y = av.y;
        b0.x = bv0.x; b0.y = bv0.y;
        b1.x = bv1.x; b1.y = bv1.y;
        c0 = __builtin_amdgcn_wmma_f32_16x16x4_f32(false, a, false, b0,
                                                   (short)0, c0, false, false);
        c1 = __builtin_amdgcn_wmma_f32_16x16x4_f32(false, a, false, b1,
                                                   (short)0, c1, false, false);
    }
    float bias0 = rb[m];
    float bias1 = rb[m + 16];

    // Per-row argmax. C layout: VGPR i -> row i (lanes 0-15) / row i+8 (lanes 16-31).
#pragma unroll
    for (int i = 0; i < 8; ++i) {
        float v0 = c0[i] + bias0;
        float v1 = c1[i] + bias1;
        float bv = v0; int be = m;                 // tie -> lower index (m < m+16)
        if (v1 > bv) { bv = v1; be = m + 16; }
        for (int off = 1; off < 16; off <<= 1) {   // stays within each 16-lane half
            float ov = __shfl_xor(bv, off, 32);
            int   oe = __shfl_xor(be, off, 32);
            if (ov > bv || (ov == bv && oe < be)) { bv = ov; be = oe; }
        }
        if (m == 0) {                              // lane 0 -> row i, lane 16 -> row i+8
            int row = i + 8 * half;
            idx[n0 + row] = be;
            atomicAdd(&counts[be], 1);
        }
    }
}

// ---------------- kernel 2: scan counts -> tile-aligned segment bases + tile map ----------------
__global__ void moe_scan(const int* __restrict__ counts,
                         int* __restrict__ baseA,
                         int* __restrict__ cursor,
                         int* __restrict__ tileExp,
                         int* __restrict__ perm) {
    for (int i = threadIdx.x; i < MAXTILES; i += blockDim.x) tileExp[i] = -1;
    for (int i = threadIdx.x; i < 16 * MAXTILES; i += blockDim.x) perm[i] = -1;
    __syncthreads();
    if (threadIdx.x == 0) {
        int ts = 0;
        for (int e = 0; e < Em; ++e) {
            baseA[e] = ts * 16;
            cursor[e] = 0;
            int T = (counts[e] + 15) >> 4;
            for (int t = 0; t < T; ++t) tileExp[ts + t] = e;
            ts += T;
        }
    }
}

// ---------------- kernel 3: scatter token ids into expert segments ----------------
__global__ void moe_scatter(const int* __restrict__ idx,
                            const int* __restrict__ baseA,
                            int* __restrict__ cursor,
                            int* __restrict__ perm) {
    int n = blockIdx.x * blockDim.x + threadIdx.x;
    int e = idx[n];
    int pos = atomicAdd(&cursor[e], 1);
    perm[baseA[e] + pos] = n;
}

// ---------------- kernel 4: fused per-tile expert MLP via f32 WMMA ----------------
// One block (4 wave32 waves) per 16-token tile of a single expert.
__global__ __launch_bounds__(128) void moe_mlp(const float* __restrict__ x,
                                               const float* __restrict__ W1,
                                               const float* __restrict__ b1,
                                               const float* __restrict__ W2,
                                               const float* __restrict__ b2,
                                               const int* __restrict__ tileExp,
                                               const int* __restrict__ perm,
                                               float* __restrict__ out) {
    __shared__ int   sTok[16];
    __shared__ float sA[16 * SA_STRIDE];
    __shared__ float sH[16 * SH_STRIDE];

    int t = blockIdx.x;
    int e = tileExp[t];
    if (e < 0) return;

    int tid = threadIdx.x;
    if (tid < 16) sTok[tid] = perm[t * 16 + tid];
    __syncthreads();

    const float* W1e = W1 + (size_t)e * Dm * Hm;
    const float* W2e = W2 + (size_t)e * Hm * Dm;

    // Warm caches with the W2 panel (64KB = 512 cachelines) while GEMM1 runs.
    {
        const char* p = (const char*)W2e;
#pragma unroll
        for (int i = 0; i < 4; ++i)
            __builtin_prefetch(p + (size_t)(tid + i * 128) * 128, 0, 0);
    }

    // Stage 16x256 A tile into LDS; padded rows -> zeros.
    for (int v = tid; v < 16 * (Dm / 4); v += 128) {
        int row = v >> 6;
        int c4  = (v & 63) * 4;
        int tok = sTok[row];
        float4 val = make_float4(0.f, 0.f, 0.f, 0.f);
        if (tok >= 0) val = *(const float4*)(x + (size_t)tok * Dm + c4);
        *(float4*)(&sA[row * SA_STRIDE + c4]) = val;
    }
    __syncthreads();

    int wave = tid >> 5;
    int lane = tid & 31;
    int half = lane >> 4;       // 0: lanes 0-15, 1: lanes 16-31
    int m    = lane & 15;       // A row / C column within tile
    int kb   = half * 2;        // K sub-offset per ISA f32 A/B layout

    // ---- GEMM1: h[16x64] = A[16x256] * W1e[256x64], wave w owns H-cols [16w,16w+16)
    // Two independent accumulator chains over k halves to hide WMMA latency.
    {
        int nn = 16 * wave + m;
        v8f ca = {0.f, 0.f, 0.f, 0.f, 0.f, 0.f, 0.f, 0.f};
        v8f cb = {0.f, 0.f, 0.f, 0.f, 0.f, 0.f, 0.f, 0.f};
        for (int k0 = 0; k0 < Dm / 2; k0 += 4) {
            int ka = k0 + kb;
            int kbh = k0 + Dm / 2 + kb;
            v2f a0, b0, a1, b1;
            a0.x = sA[m * SA_STRIDE + ka];
            a0.y = sA[m * SA_STRIDE + ka + 1];
            b0.x = W1e[(size_t)ka * Hm + nn];
            b0.y = W1e[(size_t)(ka + 1) * Hm + nn];
            a1.x = sA[m * SA_STRIDE + kbh];
            a1.y = sA[m * SA_STRIDE + kbh + 1];
            b1.x = W1e[(size_t)kbh * Hm + nn];
            b1.y = W1e[(size_t)(kbh + 1) * Hm + nn];
            ca = __builtin_amdgcn_wmma_f32_16x16x4_f32(false, a0, false, b0,
                                                       (short)0, ca, false, false);
            cb = __builtin_amdgcn_wmma_f32_16x16x4_f32(false, a1, false, b1,
                                                       (short)0, cb, false, false);
        }
        float bias = b1[e * Hm + nn];
#pragma unroll
        for (int i = 0; i < 8; ++i) {
            int row = i + 8 * half;           // C layout: VGPR i -> rows i / i+8
            float hv = ca[i] + cb[i] + bias;
            hv = hv > 0.f ? hv : 0.f;         // ReLU
            sH[row * SH_STRIDE + nn] = hv;
        }
    }
    __syncthreads();

    // ---- GEMM2: out[16x256] = h[16x64] * W2e[64x256], wave w owns D-cols [64w,64w+64)
    // 4 independent accumulator chains (one per 16-col N-tile).
    {
        v8f c2[4];
#pragma unroll
        for (int tt = 0; tt < 4; ++tt)
            c2[tt] = (v8f){0.f, 0.f, 0.f, 0.f, 0.f, 0.f, 0.f, 0.f};

        for (int k0 = 0; k0 < Hm; k0 += 4) {
            v2f a;
            a.x = sH[m * SH_STRIDE + k0 + kb];
            a.y = sH[m * SH_STRIDE + k0 + kb + 1];
#pragma unroll
            for (int tt = 0; tt < 4; ++tt) {
                int col = 64 * wave + 16 * tt + m;
                v2f b;
                b.x = W2e[(size_t)(k0 + kb) * Dm + col];
                b.y = W2e[(size_t)(k0 + kb + 1) * Dm + col];
                c2[tt] = __builtin_amdgcn_wmma_f32_16x16x4_f32(false, a, false, b,
                                                               (short)0, c2[tt], false, false);
            }
        }
#pragma unroll
        for (int tt = 0; tt < 4; ++tt) {
            int col = 64 * wave + 16 * tt + m;
            float bias = b2[e * Dm + col];
#pragma unroll
            for (int i = 0; i < 8; ++i) {
                int row = i + 8 * half;
                int tok = sTok[row];
                if (tok >= 0) {
                    float o = c2[tt][i] + bias;
                    out[(size_t)tok * Dm + col] = o > 0.f ? o : 0.f;
                }
            }
        }
    }
}

// ---------------- host launcher ----------------
extern "C" void kernel_launch(void* const* d_in, const int* in_sizes, int n_in,
                              void* d_out, int out_size, void* d_ws, size_t ws_size,
                              hipStream_t stream) {
    const float* x  = (const float*)d_in[0];   // [B,K,D] = [8192,256] flat
    const float* rw = (const float*)d_in[1];   // [E,D]
    const float* rb = (const float*)d_in[2];   // [E]
    const float* W1 = (const float*)d_in[3];   // [E,D,H]
    const float* b1 = (const float*)d_in[4];   // [E,H]
    const float* W2 = (const float*)d_in[5];   // [E,H,D]
    const float* b2 = (const float*)d_in[6];   // [E,D]
    float* out = (float*)d_out;

    int* ws      = (int*)d_ws;
    int* idx     = ws;                      // 8192
    int* counts  = idx + N_TOK;             // 32
    int* baseA   = counts + Em;             // 32
    int* cursor  = baseA + Em;              // 32
    int* tileExp = cursor + Em;             // MAXTILES
    int* perm    = tileExp + MAXTILES;      // 16*MAXTILES

    moe_zero_counts<<<1, 32, 0, stream>>>(counts);
    moe_router<<<N_TOK / 64, 128, 0, stream>>>(x, rw, rb, idx, counts);
    moe_scan<<<1, 128, 0, stream>>>(counts, baseA, cursor, tileExp, perm);
    moe_scatter<<<N_TOK / 256, 256, 0, stream>>>(idx, baseA, cursor, perm);
    moe_mlp<<<MAXTILES, 128, 0, stream>>>(x, W1, b1, W2, b2, tileExp, perm, out);
}